// LSTMAttentionDecoder_46815143526964
// MI455X (gfx1250) — compile-verified
//
#include <hip/hip_runtime.h>
#include <hip/hip_bf16.h>
#include <math.h>

// Problem constants (match reference)
#define BATCH   128
#define HDIM    512        // H_ENC == H_DEC == 512
#define NNODES  65536      // 128 trees * 512 nodes
#define NPT     512        // nodes per tree
#define VOCAB   32000
#define MTILES  (BATCH / 16)   // 8 M-tiles of 16 rows = full batch per wave

typedef __attribute__((ext_vector_type(2))) float v2f;
typedef __attribute__((ext_vector_type(8))) float v8f;

__device__ __forceinline__ float sigmoidf_(float x) { return 1.0f / (1.0f + __expf(-x)); }

// ---------------------------------------------------------------------------
// Full-M-strip fp32 WMMA GEMM (M = 128 fixed):
//   D[128 x N] = A[128 x K] * B (+ A2[128 x K2] * B2 if FUSE2) + bias0 + bias1
//   BT = true : weight stored as W[N][K] (torch Linear layout, x @ W.T)
//   BT = false: weight stored as W[K][N] (plain matmul)
// One wave per 16-column output strip; the wave iterates all 8 M-tiles so
// each weight element is streamed from HBM exactly ONCE (A is tiny and stays
// in L2). B fragment loaded once per k-step, reused by 8 WMMAs.
// blockDim.x = 128 (4 waves, 4 strips per block). N % 16 == 0, K % 4 == 0.
// Uses V_WMMA_F32_16X16X4_F32 (CDNA5, wave32).
// ---------------------------------------------------------------------------
template <bool BT, bool FUSE2>
__global__ __launch_bounds__(128)
void wmma_gemm_strip(const float* __restrict__ A,  const float* __restrict__ W,
                     const float* __restrict__ A2, const float* __restrict__ W2,
                     const float* __restrict__ bias0, const float* __restrict__ bias1,
                     float* __restrict__ D,
                     int K, int lda, int ldb,
                     int K2, int lda2, int ldb2, int ldd) {
    const int lane = threadIdx.x & 31;
    const int wave = threadIdx.x >> 5;
    const int lm   = lane & 15;      // column within strip (B/C/D N index)
    const int half = lane >> 4;      // selects K pair {0,1} vs {2,3}
    const int n    = (blockIdx.x * 4 + wave) * 16 + lm;

    v8f acc[MTILES];
#pragma unroll
    for (int mt = 0; mt < MTILES; ++mt) { v8f z = {}; acc[mt] = z; }

    // ---- first operand pair: A[128 x K] x W ----
    for (int k0 = 0; k0 < K; k0 += 4) {
        v2f b;
        if (BT) {
            b = *reinterpret_cast<const v2f*>(W + (size_t)n * ldb + k0 + 2 * half);
            __builtin_prefetch(W + (size_t)n * ldb + k0 + 64, 0, 0);  // global_prefetch_b8
        } else {
            b.x = W[(size_t)(k0 + 2 * half)     * ldb + n];
            b.y = W[(size_t)(k0 + 2 * half + 1) * ldb + n];
        }
#pragma unroll
        for (int mt = 0; mt < MTILES; ++mt) {
            v2f a = *reinterpret_cast<const v2f*>(
                A + (size_t)(mt * 16 + lm) * lda + k0 + 2 * half);
            acc[mt] = __builtin_amdgcn_wmma_f32_16x16x4_f32(
                false, a, false, b, (short)0, acc[mt], false, false);
        }
    }

    // ---- optional fused second operand pair: A2[128 x K2] x W2 ----
    if (FUSE2) {
        for (int k0 = 0; k0 < K2; k0 += 4) {
            v2f b;
            if (BT) {
                b = *reinterpret_cast<const v2f*>(W2 + (size_t)n * ldb2 + k0 + 2 * half);
                __builtin_prefetch(W2 + (size_t)n * ldb2 + k0 + 64, 0, 0);
            } else {
                b.x = W2[(size_t)(k0 + 2 * half)     * ldb2 + n];
                b.y = W2[(size_t)(k0 + 2 * half + 1) * ldb2 + n];
            }
#pragma unroll
            for (int mt = 0; mt < MTILES; ++mt) {
                v2f a = *reinterpret_cast<const v2f*>(
                    A2 + (size_t)(mt * 16 + lm) * lda2 + k0 + 2 * half);
                acc[mt] = __builtin_amdgcn_wmma_f32_16x16x4_f32(
                    false, a, false, b, (short)0, acc[mt], false, false);
            }
        }
    }

    float badd = 0.0f;
    if (bias0) badd += bias0[n];
    if (bias1) badd += bias1[n];
#pragma unroll
    for (int mt = 0; mt < MTILES; ++mt) {
#pragma unroll
        for (int r = 0; r < 8; ++r)
            D[(size_t)(mt * 16 + r + 8 * half) * ldd + n] = acc[mt][r] + badd;
    }
}

// ---------------------------------------------------------------------------
// Embedding gather: x[b, 0:512] = emb_table[token[b]]
// ---------------------------------------------------------------------------
__global__ void k_embed(const int* __restrict__ token,
                        const float* __restrict__ emb,
                        float* __restrict__ x) {
    const int b = blockIdx.x;
    const int t = threadIdx.x;            // 256 threads
    const float* row = emb + (size_t)token[b] * HDIM;
    x[(size_t)b * (2 * HDIM) + t]       = row[t];
    x[(size_t)b * (2 * HDIM) + t + 256] = row[t + 256];
}

// ---------------------------------------------------------------------------
// Per-node attention score: scores[n] = dot(enc[n], proj[seg[n]])
// One wave per node; 256-thread block handles 8 nodes.
// ---------------------------------------------------------------------------
__global__ void k_scores(const float* __restrict__ enc,
                         const float* __restrict__ proj,
                         const int* __restrict__ seg,
                         float* __restrict__ scores) {
    const int wave = threadIdx.x >> 5;
    const int lane = threadIdx.x & 31;
    const int node = blockIdx.x * 8 + wave;
    const float* e = enc  + (size_t)node * HDIM;
    const float* p = proj + (size_t)seg[node] * HDIM;
    float s = 0.0f;
#pragma unroll 4
    for (int j = lane; j < HDIM; j += 32) s += e[j] * p[j];
#pragma unroll
    for (int o = 16; o > 0; o >>= 1) s += __shfl_down(s, o, 32);
    if (lane == 0) scores[node] = s;
}

// ---------------------------------------------------------------------------
// Per-tree softmax over 512 scores. One 256-thread block per tree.
// ---------------------------------------------------------------------------
__global__ void k_softmax(const float* __restrict__ scores,
                          float* __restrict__ attn) {
    __shared__ float red[256];
    const int tree = blockIdx.x;
    const int t = threadIdx.x;
    const size_t base = (size_t)tree * NPT;

    float s0 = scores[base + t];
    float s1 = scores[base + t + 256];

    red[t] = fmaxf(s0, s1);
    __syncthreads();
    for (int o = 128; o > 0; o >>= 1) {
        if (t < o) red[t] = fmaxf(red[t], red[t + o]);
        __syncthreads();
    }
    const float mx = red[0];
    __syncthreads();

    const float e0 = __expf(s0 - mx);
    const float e1 = __expf(s1 - mx);
    red[t] = e0 + e1;
    __syncthreads();
    for (int o = 128; o > 0; o >>= 1) {
        if (t < o) red[t] += red[t + o];
        __syncthreads();
    }
    const float inv = 1.0f / red[0];

    attn[base + t]       = e0 * inv;
    attn[base + t + 256] = e1 * inv;
}

// ---------------------------------------------------------------------------
// attended[tree] = sum_n attn[tree,n] * enc[tree,n,:]  -> x[b, 512:1024]
// One 256-thread block per tree; each thread owns 2 columns; rows are read
// coalesced across the block (enc stays hot in L2 from the scores pass).
// ---------------------------------------------------------------------------
__global__ void k_attended(const float* __restrict__ enc,
                           const float* __restrict__ attn,
                           float* __restrict__ x) {
    const int tree = blockIdx.x;
    const int t = threadIdx.x;
    float a0 = 0.0f, a1 = 0.0f;
    const size_t base = (size_t)tree * NPT;
    for (int n = 0; n < NPT; ++n) {
        const float w = attn[base + n];
        const float* row = enc + (base + n) * HDIM;
        a0 += w * row[t];
        a1 += w * row[t + 256];
    }
    x[(size_t)tree * (2 * HDIM) + HDIM + t]       = a0;
    x[(size_t)tree * (2 * HDIM) + HDIM + t + 256] = a1;
}

// ---------------------------------------------------------------------------
// LSTM pointwise (torch gate order i,f,g,o): c = sig(f)*c0 + sig(i)*tanh(g);
// h = sig(o)*tanh(c). One 256-thread block per batch row, 2 units per thread.
// ---------------------------------------------------------------------------
__global__ void k_lstm(const float* __restrict__ gates,
                       const float* __restrict__ c0,
                       float* __restrict__ hout,
                       float* __restrict__ cout) {
    const int b = blockIdx.x;
    const int t = threadIdx.x;
    const size_t gb = (size_t)b * (4 * HDIM);
#pragma unroll
    for (int j = 0; j < 2; ++j) {
        const int u = t + j * 256;
        const float gi = gates[gb + u];
        const float gf = gates[gb + HDIM + u];
        const float gg = gates[gb + 2 * HDIM + u];
        const float go = gates[gb + 3 * HDIM + u];
        const float c = sigmoidf_(gf) * c0[(size_t)b * HDIM + u] +
                        sigmoidf_(gi) * tanhf(gg);
        const float h = sigmoidf_(go) * tanhf(c);
        hout[(size_t)b * HDIM + u] = h;
        cout[(size_t)b * HDIM + u] = c;
    }
}

// ---------------------------------------------------------------------------
extern "C" void kernel_launch(void* const* d_in, const int* in_sizes, int n_in,
                              void* d_out, int out_size, void* d_ws, size_t ws_size,
                              hipStream_t stream) {
    const int*   token  = (const int*)  d_in[0];
    const float* h0     = (const float*)d_in[1];   // [1,B,H] -> [B,H]
    const float* c0     = (const float*)d_in[2];
    const float* enc    = (const float*)d_in[3];   // [N,H]
    const int*   seg    = (const int*)  d_in[4];
    const float* emb    = (const float*)d_in[5];   // [V,H]
    const float* W_attn = (const float*)d_in[6];   // [H,H]   (K x N)
    const float* W_ih   = (const float*)d_in[7];   // [4H, 2H] (N x K)
    const float* W_hh   = (const float*)d_in[8];   // [4H, H]  (N x K)
    const float* b_ih   = (const float*)d_in[9];
    const float* b_hh   = (const float*)d_in[10];
    const float* W_out  = (const float*)d_in[11];  // [V, H]   (N x K)
    const float* b_out  = (const float*)d_in[12];

    float* out    = (float*)d_out;
    float* logits = out;                                   // [B, V]
    float* hout   = out + (size_t)BATCH * VOCAB;           // [B, H]
    float* cout   = hout + (size_t)BATCH * HDIM;           // [B, H]

    float* ws     = (float*)d_ws;
    float* proj   = ws;                                    // B*H
    float* scores = proj   + (size_t)BATCH * HDIM;         // N
    float* attn   = scores + (size_t)NNODES;               // N
    float* x      = attn   + (size_t)NNODES;               // B * 2H
    float* gates  = x      + (size_t)BATCH * 2 * HDIM;     // B * 4H

    // 1. proj = h0 @ W_attn    [128,512] = [128,512] x [512,512](KxN)
    wmma_gemm_strip<false, false><<<HDIM / 64, 128, 0, stream>>>(
        h0, W_attn, nullptr, nullptr, nullptr, nullptr, proj,
        HDIM, HDIM, HDIM, 0, 0, 0, HDIM);

    // 2. x[:, :512] = emb_table[token]   (independent of proj)
    k_embed<<<BATCH, 256, 0, stream>>>(token, emb, x);

    // 3. per-node scores
    k_scores<<<NNODES / 8, 256, 0, stream>>>(enc, proj, seg, scores);

    // 4. per-tree softmax
    k_softmax<<<BATCH, 256, 0, stream>>>(scores, attn);

    // 5. attended -> x[:, 512:]
    k_attended<<<BATCH, 256, 0, stream>>>(enc, attn, x);

    // 6. gates = x @ W_ih.T + h0 @ W_hh.T + b_ih + b_hh   (fused, [128,2048])
    wmma_gemm_strip<true, true><<<(4 * HDIM) / 64, 128, 0, stream>>>(
        x, W_ih, h0, W_hh, b_ih, b_hh, gates,
        2 * HDIM, 2 * HDIM, 2 * HDIM,          // K=1024 over x / W_ih
        HDIM, HDIM, HDIM,                      // K2=512 over h0 / W_hh
        4 * HDIM);

    // 7. LSTM pointwise -> hout, cout
    k_lstm<<<BATCH, 256, 0, stream>>>(gates, c0, hout, cout);

    // 8. logits = h @ W_out.T + b_out   [128,32000], K=512, W_out read ONCE
    wmma_gemm_strip<true, false><<<VOCAB / 64, 128, 0, stream>>>(
        hout, W_out, nullptr, nullptr, b_out, nullptr, logits,
        HDIM, HDIM, HDIM, 0, 0, 0, VOCAB);
}